// Sinkhorn_6339371729578
// MI455X (gfx1250) — compile-verified
//
#include <hip/hip_runtime.h>
#include <math.h>

// Sinkhorn [32,1024,1024], TAU=1, 10 iters.
// Rank-1 reformulation: log_s = L - R_i - C_j, alternate
//   R_i = logsumexp_j(L - C_j),  C_j = logsumexp_i(L - R_i),
// out = exp(L - R - C). Only the 128MB input is streamed (L2-resident after
// pass 1); intermediates are 2x 128KB vectors in d_ws.

typedef __attribute__((ext_vector_type(2))) float v2f;
typedef __attribute__((ext_vector_type(8))) float v8f;

#define NB 32
#define N  1024

// ---------------------------------------------------------------------------
// Row pass: one wave owns a 16-row x 1024-col tile.
// Pass1: per-row max of (s - C). Pass2: sum exp via V_WMMA_F32_16X16X4_F32
// with B = ones (D += A x 1 accumulates exact f32 row sums on the matrix pipe,
// co-executing with v_exp_f32 on the trans pipe).
// A-matrix f32 16x4 layout: lane l -> row l%16; VGPR0/1 hold K = 2*(l>>4)+{0,1}.
// ---------------------------------------------------------------------------
template <bool USE_C>
__global__ __launch_bounds__(256) void sinkhorn_row(const float* __restrict__ s,
                                                    const float* __restrict__ C,
                                                    float* __restrict__ R)
{
    const int lane = threadIdx.x & 31;
    const int wave = threadIdx.x >> 5;
    const int tile = blockIdx.x * 8 + wave;   // 32 * 64 = 2048 tiles
    const int b    = tile >> 6;
    const int row0 = (tile & 63) << 4;
    const int r    = lane & 15;
    const int half = lane >> 4;

    const float* srow = s + ((size_t)(b * N + row0 + r)) * N;
    const float* crow = C + (size_t)b * N;

    // ---- pass 1: per-row max of (s - C), stabilized logsumexp ----
    float m = -INFINITY;
    #pragma unroll 4
    for (int t = 0; t < N; t += 4) {
        const int col = t + 2 * half;
        v2f x = *(const v2f*)(srow + col);
        if (USE_C) {
            v2f cc = *(const v2f*)(crow + col);
            x.x -= cc.x; x.y -= cc.y;
        }
        m = fmaxf(m, fmaxf(x.x, x.y));
    }
    // combine the two K-halves of each row (lane r <-> lane r+16)
    m = fmaxf(m, __shfl_xor(m, 16, 32));

    // ---- pass 2: accumulate exp(x - m) row sums with WMMA (L2-resident reread)
    v2f ones; ones.x = 1.0f; ones.y = 1.0f;   // all-ones B: layout-invariant
    v8f acc = {};
    #pragma unroll 2
    for (int t = 0; t < N; t += 4) {
        const int col = t + 2 * half;
        v2f x = *(const v2f*)(srow + col);
        if (USE_C) {
            v2f cc = *(const v2f*)(crow + col);
            x.x -= cc.x; x.y -= cc.y;
        }
        v2f a;
        a.x = __expf(x.x - m);
        a.y = __expf(x.y - m);
        // (neg_a, A, neg_b, B, c_mod, C, reuse_a, reuse_b)
        acc = __builtin_amdgcn_wmma_f32_16x16x4_f32(false, a, false, ones,
                                                    (short)0, acc, false, false);
    }

    // D layout: element (M,Ncol) lives in VGPR M&7, lane half M>>3, Ncol = lane%16.
    // Lane with (l%16 == M) also holds row M's max -> lanes 0..7 write rows 0..7,
    // lanes 24..31 write rows 8..15.
    if (half == (r >> 3)) {
        R[(size_t)b * N + row0 + r] = m + __logf(acc[r & 7]);
    }
}

// ---------------------------------------------------------------------------
// Col pass: thread-per-column, fully coalesced (32 lanes x 4B per row step),
// R_i is wave-uniform (scalar broadcast). Single-pass online logsumexp:
// the rescale branch fires only when the running max changes (~log N times).
// ---------------------------------------------------------------------------
__global__ __launch_bounds__(64) void sinkhorn_col(const float* __restrict__ s,
                                                   const float* __restrict__ R,
                                                   float* __restrict__ C)
{
    const int b = blockIdx.x >> 4;                       // N/64 = 16 blocks/matrix
    const int j = ((blockIdx.x & 15) << 6) + threadIdx.x;
    const float* sm = s + (size_t)b * N * N + j;
    const float* Rb = R + (size_t)b * N;

    float m = -INFINITY, acc = 0.0f;
    for (int i = 0; i < N; ++i) {
        const float x = sm[(size_t)i * N] - Rb[i];
        if (x > m) { acc *= __expf(m - x); m = x; }
        acc += __expf(x - m);
    }
    C[(size_t)b * N + j] = m + __logf(acc);
}

// ---------------------------------------------------------------------------
// Final pass: out = exp(s - R_i - C_j), float4 vectorized (b128 loads/stores).
// ---------------------------------------------------------------------------
__global__ __launch_bounds__(256) void sinkhorn_final(const float* __restrict__ s,
                                                      const float* __restrict__ R,
                                                      const float* __restrict__ C,
                                                      float* __restrict__ out)
{
    const size_t e0 = ((size_t)blockIdx.x * 256 + threadIdx.x) * 4;
    const int b   = (int)(e0 >> 20);         // / (N*N)
    const int rem = (int)(e0 & (N * N - 1));
    const int i   = rem >> 10;
    const int j   = rem & (N - 1);

    const float  r  = R[b * N + i];
    const float4 sv = *(const float4*)(s + e0);
    const float4 cv = *(const float4*)(C + (size_t)b * N + j);
    float4 o;
    o.x = __expf(sv.x - r - cv.x);
    o.y = __expf(sv.y - r - cv.y);
    o.z = __expf(sv.z - r - cv.z);
    o.w = __expf(sv.w - r - cv.w);
    *(float4*)(out + e0) = o;
}

extern "C" void kernel_launch(void* const* d_in, const int* in_sizes, int n_in,
                              void* d_out, int out_size, void* d_ws, size_t ws_size,
                              hipStream_t stream)
{
    (void)in_sizes; (void)n_in; (void)out_size; (void)ws_size;
    const float* s = (const float*)d_in[0];
    float* out = (float*)d_out;
    float* R = (float*)d_ws;           // [NB*N] f32
    float* C = R + NB * N;             // [NB*N] f32

    const int ROW_BLOCKS = (NB * (N / 16)) / 8;   // 256 blocks x 8 waves
    const int COL_BLOCKS = NB * (N / 64);         // 512 blocks x 2 waves
    const int FIN_BLOCKS = (NB * N * N / 4) / 256;

    // iter 0: row pass with C == 0 (reads nothing from workspace)
    sinkhorn_row<false><<<ROW_BLOCKS, 256, 0, stream>>>(s, nullptr, R);
    for (int it = 1; it < 10; ++it) {
        if (it & 1) sinkhorn_col<<<COL_BLOCKS, 64, 0, stream>>>(s, R, C);
        else        sinkhorn_row<true><<<ROW_BLOCKS, 256, 0, stream>>>(s, C, R);
    }
    sinkhorn_final<<<FIN_BLOCKS, 256, 0, stream>>>(s, R, C, out);
}